// EMRouting_70617852280850
// MI455X (gfx1250) — compile-verified
//
#include <hip/hip_runtime.h>

// ---------------- problem constants (from reference) ----------------
#define BB      8
#define HH      12
#define WWD     12
#define KKW     3
#define CIN     16
#define CO      32          // COUT
#define PD      16          // P
#define NN      144         // K*K*CIN votes per location
#define NC      512         // CO*PD columns
#define SLAB    (NN * NC)   // 73728 floats = 288 KB per location
#define NLOC    1152        // B*H*W
#define NTHREADS 1024
#define EPSF     1e-7f
#define NITER    3
#define LOG2PIF  1.8378770664093453f   // log(2*pi)

#define MU_OFF  0
#define A_OFF   ((size_t)NLOC * NC)             // 589824
#define SIG_OFF (A_OFF + (size_t)NLOC * CO)     // 626688

typedef float v2f  __attribute__((ext_vector_type(2)));
typedef float v8f  __attribute__((ext_vector_type(8)));
typedef unsigned int u32x4 __attribute__((ext_vector_type(4)));
typedef int   i32x4 __attribute__((ext_vector_type(4)));
typedef int   i32x8 __attribute__((ext_vector_type(8)));

// async-load builtin parameter types: pointer to int __attribute__((vector_size(16)))
typedef int v4i __attribute__((vector_size(4 * sizeof(int))));
typedef __attribute__((address_space(1))) v4i g_v4i;   // global
typedef __attribute__((address_space(3))) v4i l_v4i;   // LDS

// ---- CDNA5 feature detection (guards keep host pass + older clangs compiling) ----
#if __has_builtin(__builtin_amdgcn_global_load_async_to_lds_b128)
#define USE_ASYNC_B128 1
#elif __has_builtin(__builtin_amdgcn_tensor_load_to_lds)
#define USE_TDM 1
#endif
#if __has_builtin(__builtin_amdgcn_wmma_f32_16x16x4_f32)
#define USE_WMMA 1
#endif

__global__ __launch_bounds__(NTHREADS, 1)
void em_routing_fused(const float* __restrict__ gV,
                      const float* __restrict__ gA,
                      const float* __restrict__ gBu,
                      const float* __restrict__ gBa,
                      const float* __restrict__ gR,
                      float* __restrict__ gOut)
{
    // ~314 KB static LDS: fits the 320 KB CDNA5 WGP budget -> 1 WG/WGP, 32 waves
    __shared__ __align__(16) float sV[SLAB];     // V slab [n][c][p]
    __shared__ float sRw[NN * CO];               // R*a (iteration state)
    __shared__ float sA[NN];
    __shared__ float sMu[NC];
    __shared__ float sSig[NC];
    __shared__ float sInv2S[NC];
    __shared__ float sX[CO];
    __shared__ float sLsum[CO];
    __shared__ float sAout[CO];
    __shared__ float sLg[CO];
    __shared__ float sRowMax[NN];
    __shared__ float sRowScale[NN];

    const int tid = threadIdx.x;
    const int loc = blockIdx.x;
    const float* __restrict__ vbase = gV + (size_t)loc * SLAB;

    // ---------- stage V slab into LDS: the ONLY HBM read of V ----------
#if defined(USE_ASYNC_B128)
    for (int i = tid; i < SLAB / 4; i += NTHREADS) {
        __builtin_amdgcn_global_load_async_to_lds_b128(
            (g_v4i*)(vbase + 4 * i), (l_v4i*)(&sV[4 * i]), 0, 0);
    }
#elif defined(USE_TDM)
    if (tid < 32) {   // one wave issues one TDM descriptor: 2D tile 512 x 144 f32
        unsigned long long ga64 = (unsigned long long)(uintptr_t)vbase;
        unsigned int lds_off = (unsigned int)(uintptr_t)(void*)&sV[0];
        u32x4 g0 = { 1u,                                   // count=1
                     lds_off,                              // lds_addr
                     (unsigned int)(ga64 & 0xFFFFFFFFull), // global_addr[31:0]
                     (unsigned int)((ga64 >> 32) & 0x1FFFFFFull) | (2u << 30) }; // [56:32] | type=2
        i32x8 g1 = { (int)0x00020000,                                  // data_size=4B
                     (int)(((unsigned)NC & 0xFFFFu) << 16),            // tensor_dim0 lo
                     (int)((((unsigned)NC >> 16) & 0xFFFFu) | (((unsigned)NN & 0xFFFFu) << 16)),
                     (int)((((unsigned)NN >> 16) & 0xFFFFu) | (((unsigned)NC & 0xFFFFu) << 16)), // tile_dim0
                     (int)NN,                                          // tile_dim1 (tile_dim2=0)
                     (int)NC,                                          // tensor_dim0_stride lo
                     0, 0 };
        i32x4 z4 = { 0, 0, 0, 0 };
#if defined(__clang_major__) && __clang_major__ >= 23
        i32x8 z8 = { 0, 0, 0, 0, 0, 0, 0, 0 };
        __builtin_amdgcn_tensor_load_to_lds(g0, g1, z4, z4, z8, 0);
#else
        __builtin_amdgcn_tensor_load_to_lds(g0, g1, z4, z4, 0);
#endif
    }
#else
    for (int i = tid; i < SLAB / 4; i += NTHREADS)
        ((float4*)sV)[i] = ((const float4*)vbase)[i];
#endif

    // overlap with the async copy: load a, build Rw = R*a
    if (tid < NN) sA[tid] = gA[(size_t)loc * NN + tid];
    __syncthreads();
    {
        const float* __restrict__ rbase = gR + (size_t)loc * (NN * CO);
        for (int i = tid; i < NN * CO; i += NTHREADS)
            sRw[i] = rbase[i] * sA[i >> 5];
    }

#if defined(USE_ASYNC_B128)
#if __has_builtin(__builtin_amdgcn_s_wait_asynccnt)
    __builtin_amdgcn_s_wait_asynccnt(0);
#else
    asm volatile("s_wait_asynccnt 0" ::: "memory");
#endif
#elif defined(USE_TDM)
    if (tid < 32) {
#if __has_builtin(__builtin_amdgcn_s_wait_tensorcnt)
        __builtin_amdgcn_s_wait_tensorcnt(0);
#else
        asm volatile("s_wait_tensorcnt 0" ::: "memory");
#endif
    }
#endif
    __syncthreads();

    // wave g owns output capsule (cout) g; lane layout for f32 WMMA B operand
    const int g    = tid >> 5;          // 0..31
    const int lane = tid & 31;
    const int p    = lane & 15;
    const int koff = (lane >> 4) << 1;  // K row 0/2 per half-wave (1/3 via b1)

    float att = 1.0f;
    for (int iter = 0; iter < NITER; ++iter) {
        att *= 0.95f;
        const float Lam = 0.01f * (1.0f - att);

        // ---- mu_denom[g] = sum_n Rw[n,g]  (wave-local butterfly) ----
        float dsum = 0.f;
        for (int n = lane; n < NN; n += 32) dsum += sRw[n * CO + g];
#pragma unroll
        for (int m = 16; m >= 1; m >>= 1) dsum += __shfl_xor(dsum, m, 32);
        const float den  = dsum + EPSF;
        const float dinv = 1.0f / den;

        // ---- mu[g,p] = sum_n Rw[n,g]*V[n,g,p] : WMMA ones-reduction ----
#if defined(USE_WMMA)
        const v2f ones = { 1.0f, 1.0f };
        v8f acc = {};
        for (int s = 0; s < NN / 4; ++s) {
            const int n0 = 4 * s + koff;
            float b0 = sRw[n0 * CO + g]       * sV[n0 * NC + g * PD + p];
            float b1 = sRw[(n0 + 1) * CO + g] * sV[(n0 + 1) * NC + g * PD + p];
            v2f bv = { b0, b1 };
            acc = __builtin_amdgcn_wmma_f32_16x16x4_f32(
                      false, ones, false, bv, (short)0, acc, false, false);
        }
        const float mu_c = acc[0] * dinv;   // all D rows equal the column sum
#else
        float part = 0.f;
        const int nb = (lane >> 4) * 72;
        for (int n = nb; n < nb + 72; ++n)
            part += sRw[n * CO + g] * sV[n * NC + g * PD + p];
        part += __shfl_xor(part, 16, 32);
        const float mu_c = part * dinv;
#endif
        if (lane < 16) sMu[g * PD + p] = mu_c;

        // ---- sigma_sqr[g,p] = sum_n Rw*(V-mu)^2 / den ----
#if defined(USE_WMMA)
        v8f acc2 = {};
        for (int s = 0; s < NN / 4; ++s) {
            const int n0 = 4 * s + koff;
            float d0 = sV[n0 * NC + g * PD + p]       - mu_c;
            float d1 = sV[(n0 + 1) * NC + g * PD + p] - mu_c;
            v2f bv = { sRw[n0 * CO + g] * d0 * d0,
                       sRw[(n0 + 1) * CO + g] * d1 * d1 };
            acc2 = __builtin_amdgcn_wmma_f32_16x16x4_f32(
                       false, ones, false, bv, (short)0, acc2, false, false);
        }
        const float sig = acc2[0] * dinv;
#else
        float part2 = 0.f;
        for (int n = nb; n < nb + 72; ++n) {
            float d = sV[n * NC + g * PD + p] - mu_c;
            part2 += sRw[n * CO + g] * d * d;
        }
        part2 += __shfl_xor(part2, 16, 32);
        const float sig = part2 * dinv;
#endif
        if (lane < 16) {
            sSig[g * PD + p]   = sig;
            sInv2S[g * PD + p] = 1.0f / (2.0f * sig + EPSF);
        }

        // sum_p log(sigma) within 16-lane group (halves hold identical data)
        float lsum = __logf(sig);
#pragma unroll
        for (int m = 8; m >= 1; m >>= 1) lsum += __shfl_xor(lsum, m, 32);

        if (lane == 0) {
            sLsum[g] = lsum;
            const float bu = gBu[g], ba = gBa[g];
            // x = Lam * (Beta_a - sum_p ((Beta_u + 0.5 log sig) * denom))
            sX[g] = Lam * (ba - dsum * (16.0f * bu + 0.5f * lsum));
        }
        __syncthreads();

        // ---- a_out = sigmoid(l2_normalize(x over cout)); log_p1 ----
        if (tid < CO) {
            float ss = 0.f;
            for (int c = 0; c < CO; ++c) ss += sX[c] * sX[c];
            const float nrm = sqrtf(ss);
            const float xn  = sX[tid] / fmaxf(nrm, 1e-12f);
            const float ao  = 1.0f / (1.0f + __expf(-xn));
            sAout[tid] = ao;
            sLg[tid] = __logf(ao) + (-0.5f * (16.0f * LOG2PIF + sLsum[tid]) + EPSF);
        }
        __syncthreads();

        // ---- R update (softmax over cout), skipped on the last iteration ----
        if (iter < NITER - 1) {
            for (int idx = tid; idx < NN * CO; idx += NTHREADS) {
                const int c = idx / NN, n = idx - c * NN;   // n-minor for LDS banks
                const float* __restrict__ vp = &sV[n * NC + c * PD];
                const float* __restrict__ mp = &sMu[c * PD];
                const float* __restrict__ ip = &sInv2S[c * PD];
                float s2 = 0.f;
#pragma unroll
                for (int q = 0; q < PD; ++q) {
                    const float d = vp[q] - mp[q];
                    s2 += d * d * ip[q];
                }
                sRw[n * CO + c] = sLg[c] - s2;   // logits overwrite Rw in place
            }
            __syncthreads();
            if (tid < NN) {
                float m = -3.0e38f;
                for (int c = 0; c < CO; ++c) m = fmaxf(m, sRw[tid * CO + c]);
                float se = 0.f;
                for (int c = 0; c < CO; ++c) se += __expf(sRw[tid * CO + c] - m);
                sRowMax[tid]   = m;
                sRowScale[tid] = sA[tid] / se;   // fold a[n] into next Rw
            }
            __syncthreads();
            for (int idx = tid; idx < NN * CO; idx += NTHREADS) {
                const int n = idx >> 5;
                sRw[idx] = __expf(sRw[idx] - sRowMax[n]) * sRowScale[n];
            }
            __syncthreads();
        }
    }

    // ---- outputs: mu | a_out | sigma, concatenated flat ----
    if (tid < NC) {
        gOut[MU_OFF  + (size_t)loc * NC + tid] = sMu[tid];
        gOut[SIG_OFF + (size_t)loc * NC + tid] = sSig[tid];
    }
    if (tid < CO)
        gOut[A_OFF + (size_t)loc * CO + tid] = sAout[tid];
}

extern "C" void kernel_launch(void* const* d_in, const int* in_sizes, int n_in,
                              void* d_out, int out_size, void* d_ws, size_t ws_size,
                              hipStream_t stream) {
    (void)in_sizes; (void)n_in; (void)out_size; (void)d_ws; (void)ws_size;
    const float* V  = (const float*)d_in[0];
    const float* a  = (const float*)d_in[1];
    const float* Bu = (const float*)d_in[2];
    const float* Ba = (const float*)d_in[3];
    const float* R  = (const float*)d_in[4];
    float* out = (float*)d_out;
    em_routing_fused<<<NLOC, NTHREADS, 0, stream>>>(V, a, Bu, Ba, R, out);
}